// VDPFirstConv_89541478187433
// MI455X (gfx1250) — compile-verified
//
#include <hip/hip_runtime.h>

// Problem constants (from reference)
#define BATCH 2
#define CINCH 3
#define HIMG  48
#define KSZ   5
#define OH    44                 // 48 - 5 + 1
#define PNUM  (OH*OH)            // 1936 patches, multiple of 16
#define FEAT  75                 // CIN*K*K
#define FPAD  76                 // padded to multiple of 4 for 16x16x4 WMMA K-steps
#define KNUM  16
#define NTILE (PNUM/16)          // 121 tiles per dim
#define TILES_PER_B (NTILE*NTILE)    // 14641
#define NTILES (BATCH*TILES_PER_B)   // 29282
#define MU_ELEMS (BATCH*KNUM*PNUM)   // 61952
#define XM_FLOATS (BATCH*PNUM*FPAD)  // 294272

typedef __attribute__((ext_vector_type(2))) float v2f;
typedef __attribute__((ext_vector_type(4))) float v4f;
typedef __attribute__((ext_vector_type(8))) float v8f;

// ---------------------------------------------------------------------------
// Kernel 0: one-shot softplus(w_sigma) -> ws (16 floats). Hoists ~2k scalar
// instructions of exp/log expansion out of every one of the 29282 hot waves.
// ---------------------------------------------------------------------------
__global__ void softplus_prep_kernel(const float* __restrict__ w_sigma,
                                     float* __restrict__ sp_out) {
    int t = threadIdx.x;
    if (t < KNUM) sp_out[t] = log1pf(expf(w_sigma[t]));
}

// ---------------------------------------------------------------------------
// Kernel 1: build padded patch matrix xm[B][P][FPAD] in workspace.
// f ordering matches reference: (c, kh, kw), c slowest; f==75 is zero pad.
// ---------------------------------------------------------------------------
__global__ void build_patches_kernel(const float* __restrict__ mu_in,
                                     float* __restrict__ xm) {
    int t = blockIdx.x * blockDim.x + threadIdx.x;
    const int total = BATCH * PNUM * FPAD;
    if (t >= total) return;
    int f = t % FPAD;
    int p = (t / FPAD) % PNUM;
    int b = t / (FPAD * PNUM);
    float v = 0.0f;
    if (f < FEAT) {
        int c  = f / (KSZ * KSZ);
        int rs = f % (KSZ * KSZ);
        int r  = rs / KSZ;
        int s  = rs % KSZ;
        int oh = p / OH;
        int ow = p % OH;
        v = mu_in[((b * CINCH + c) * HIMG + (oh + r)) * HIMG + (ow + s)];
    }
    xm[t] = v;
}

// ---------------------------------------------------------------------------
// Kernel 2: mean path conv. mu_out[b][kn][p] = xm[b][p][:] . w_mu[kn][:]
// 9.3 MFLOP total — negligible; one thread per output.
// ---------------------------------------------------------------------------
__global__ void conv_mu_kernel(const float* __restrict__ xm,
                               const float* __restrict__ w_mu,
                               float* __restrict__ mu_out) {
    int t = blockIdx.x * blockDim.x + threadIdx.x;
    if (t >= MU_ELEMS) return;
    int p  = t % PNUM;
    int kn = (t / PNUM) % KNUM;
    int b  = t / (PNUM * KNUM);
    const float* xr = xm + ((size_t)b * PNUM + p) * FPAD;
    const float* wr = w_mu + kn * FEAT;
    float acc = 0.0f;
    #pragma unroll
    for (int f = 0; f < FEAT; ++f) acc += xr[f] * wr[f];
    mu_out[t] = acc;
}

// ---------------------------------------------------------------------------
// Kernel 3: fused variance path. One wave32 per 16x16 tile of xxT = xm.xm^T,
// accumulated with V_WMMA_F32_16X16X4_F32 (19 K-steps over FPAD=76).
// Epilogue: scale by precomputed softplus(w_sigma[kn]), finite-guard, and
// stream 16 KB per tile with nontemporal stores (480 MB write-once output).
// Per-KN base pointer + constant row offsets keep address math off VALU.
// ---------------------------------------------------------------------------
__global__ void __launch_bounds__(256)
sigma_wmma_kernel(const float* __restrict__ xm,
                  const float* __restrict__ sp_pre,
                  float* __restrict__ sigma) {
    int wave = blockIdx.x * (blockDim.x >> 5) + (threadIdx.x >> 5);
    int lane = threadIdx.x & 31;
    if (wave >= NTILES) return;               // wave-uniform: EXEC stays all-1s

    int b   = wave / TILES_PER_B;
    int rem = wave % TILES_PER_B;
    int ti  = rem / NTILE;
    int tj  = rem % NTILE;

    // precomputed softplus values: 64 bytes, vector loads (hits L2)
    v4f s0 = ((const v4f*)sp_pre)[0];
    v4f s1 = ((const v4f*)sp_pre)[1];
    v4f s2 = ((const v4f*)sp_pre)[2];
    v4f s3 = ((const v4f*)sp_pre)[3];
    float sp[KNUM] = {s0.x, s0.y, s0.z, s0.w, s1.x, s1.y, s1.z, s1.w,
                      s2.x, s2.y, s2.z, s2.w, s3.x, s3.y, s3.z, s3.w};

    int half = lane >> 4;                     // 0: K pair {0,1}, 1: K pair {2,3}
    int l16  = lane & 15;
    const float* arow = xm + ((size_t)b * PNUM + (size_t)ti * 16 + l16) * FPAD + 2 * half;
    const float* brow = xm + ((size_t)b * PNUM + (size_t)tj * 16 + l16) * FPAD + 2 * half;

    v8f acc = {0.f, 0.f, 0.f, 0.f, 0.f, 0.f, 0.f, 0.f};
    #pragma unroll
    for (int kk = 0; kk < FPAD / 4; ++kk) {   // 19 WMMA steps
        v2f afrag = *(const v2f*)(arow + kk * 4);
        v2f bfrag = *(const v2f*)(brow + kk * 4);
        acc = __builtin_amdgcn_wmma_f32_16x16x4_f32(
            /*neg_a=*/false, afrag, /*neg_b=*/false, bfrag,
            /*c_mod=*/(short)0, acc, /*reuse_a=*/false, /*reuse_b=*/false);
    }

    // Epilogue. D layout: VGPR v, lanes 0-15: row ti*16+v; lanes 16-31: +8.
    // Base pointer covers (b, kn=0, row0 = ti*16 + 8*half, col); row stride
    // v*PNUM*4 = 7744 B fits the signed 24-bit immediate offset field.
    float* out0 = sigma
        + ((size_t)b * KNUM * PNUM + (size_t)ti * 16 + 8 * half) * PNUM
        + (size_t)tj * 16 + l16;
    #pragma unroll
    for (int k = 0; k < KNUM; ++k) {
        float s = sp[k];
        float* pk = out0 + (size_t)k * PNUM * PNUM;   // uniform stride per plane
        #pragma unroll
        for (int v = 0; v < 8; ++v) {
            float val = acc[v] * s;
            // reference: where(isfinite, x, 0)
            if ((__float_as_uint(val) & 0x7F800000u) == 0x7F800000u) val = 0.0f;
            __builtin_nontemporal_store(val, pk + (size_t)v * PNUM);
        }
    }
}

// ---------------------------------------------------------------------------
extern "C" void kernel_launch(void* const* d_in, const int* in_sizes, int n_in,
                              void* d_out, int out_size, void* d_ws, size_t ws_size,
                              hipStream_t stream) {
    const float* mu_in   = (const float*)d_in[0];   // [2,3,48,48]
    const float* w_mu    = (const float*)d_in[1];   // [16,3,5,5]
    const float* w_sigma = (const float*)d_in[2];   // [16]
    float* out = (float*)d_out;                     // mu_out (61952) ++ sigma

    float* xm = (float*)d_ws;                       // [B][PNUM][FPAD] = 1.18 MB
    float* sp = xm + XM_FLOATS;                     // 16 floats (16B-aligned)

    // 0) one-shot softplus
    softplus_prep_kernel<<<1, 32, 0, stream>>>(w_sigma, sp);
    // 1) patch matrix into workspace (lives in L2 for the GEMM pass)
    {
        int total = BATCH * PNUM * FPAD;
        build_patches_kernel<<<(total + 255) / 256, 256, 0, stream>>>(mu_in, xm);
    }
    // 2) mean-path conv
    conv_mu_kernel<<<(MU_ELEMS + 255) / 256, 256, 0, stream>>>(xm, w_mu, out);
    // 3) fused WMMA xxT + softplus-broadcast + NT stream-out (bandwidth-bound)
    {
        int waves_per_block = 256 / 32;
        int blocks = (NTILES + waves_per_block - 1) / waves_per_block;  // 3661
        sigma_wmma_kernel<<<blocks, 256, 0, stream>>>(xm, sp, out + MU_ELEMS);
    }
}